// SpanSequnce_9878424781362
// MI455X (gfx1250) — compile-verified
//
#include <hip/hip_runtime.h>
#include <math.h>

// Problem constants (B,S,L,H,D) = (32, 512, 8, 1024, 64)
#define BB 32
#define SS 512
#define LL 8
#define HH 1024
#define DD 64
#define ROWS (BB * SS)          // 16384 positions
#define RPW 64                  // rows per workgroup (kernel 1)
#define NTHR 128                // threads per workgroup (4 waves)
#define KC  64                  // K chunk
#define NCHUNK (HH / KC)        // 16
#define XS_STRIDE 68            // LDS stride for X tile (bank-conflict pad)
#define WT_FLOATS (KC * DD)     // 4096 floats per pre-swizzled W1 chunk

typedef __attribute__((ext_vector_type(2))) float v2f;
typedef __attribute__((ext_vector_type(8))) float v8f;

// ---------------------------------------------------------------------------
// Kernel 0 (one-off): rewrite W1 (H x D, row major) into WMMA B-fragment
// order per ISA 7.12.2 (32-bit B 4x16): for chunk c, K-step ks, N-tile n,
// lane ln holds pair { W1[c*64+ks*4+2*(ln>>4)+0][n*16+(ln&15)],
//                     W1[c*64+ks*4+2*(ln>>4)+1][n*16+(ln&15)] }.
// Flat: w1s[(((c*16+ks)*4+n)*32 + ln)*2 + j]. 32768 pairs total.
// ---------------------------------------------------------------------------
__global__ __launch_bounds__(256)
void swizzle_w1_kernel(const float* __restrict__ W1, float* __restrict__ w1s) {
    const int t  = blockIdx.x * 256 + threadIdx.x;  // 0..32767 (pair index)
    const int ln = t & 31;
    const int n  = (t >> 5) & 3;
    const int ks = (t >> 7) & 15;
    const int c  = t >> 11;
    const int k   = c * KC + ks * 4 + 2 * (ln >> 4);
    const int col = n * 16 + (ln & 15);
    v2f v;
    v.x = W1[(size_t)k * DD + col];
    v.y = W1[(size_t)(k + 1) * DD + col];
    *(v2f*)&w1s[(size_t)t * 2] = v;
}

// ---------------------------------------------------------------------------
// Kernel 1: per-position MLP e_pos[b,p] = relu((h*tw)@W1 + b1)@W2 + b2
//           and gs[b,p] = h @ Ws, using V_WMMA_F32_16X16X4_F32.
// Grid: 256 blocks x 128 threads (4 waves). Block g owns rows [g*64, g*64+64).
// Each wave computes a 16x64 C tile (4 accumulators).
// ---------------------------------------------------------------------------
__global__ __launch_bounds__(NTHR)
void span_mlp_wmma_kernel(const float* __restrict__ hidden,
                          const float* __restrict__ tw,
                          const float* __restrict__ w1s,   // pre-swizzled W1
                          const float* __restrict__ b1,
                          const float* __restrict__ W2,
                          const float* __restrict__ b2,
                          const float* __restrict__ Ws,
                          float* __restrict__ e_out,
                          float* __restrict__ gs_out) {
    __shared__ float xs[RPW * XS_STRIDE];   // 17408 B (A tiles, padded)
    __shared__ float wt[WT_FLOATS];         // 16384 B (B fragments, contiguous)
    __shared__ float gsp[RPW * 17];         //  4352 B (gs partials)

    const int tid  = threadIdx.x;
    const int quad = tid & 15;              // which float4 (k-quad) this thread loads
    const int rb   = tid >> 4;              // row base 0..7
    const int wave = tid >> 5;              // 0..3
    const int lane = tid & 31;
    const int lm   = lane & 15;             // M index within A fragment
    const int lh   = lane >> 4;             // lane half -> K pair select
    const int wbase = wave * 16;            // this wave's local row tile
    const long wg_row0 = (long)blockIdx.x * RPW;

    const float4* h4   = (const float4*)hidden;
    const float4* tw4  = (const float4*)tw;
    const float4* ws4  = (const float4*)Ws;
    const float4* w1s4 = (const float4*)w1s;

    v8f acc[4] = {};        // 16 rows x 64 cols of C, 4 N-tiles
    float gs_acc[8];
    #pragma unroll
    for (int i = 0; i < 8; ++i) gs_acc[i] = 0.f;

    for (int c = 0; c < NCHUNK; ++c) {
        const int k0 = c * KC;
        // --- stage X tile (scaled by term_weight), fuse gs = h . Ws ---
        const float4 twv = tw4[(k0 >> 2) + quad];
        const float4 wsv = ws4[(k0 >> 2) + quad];
        #pragma unroll
        for (int i = 0; i < 8; ++i) {
            const int lr = rb + i * 8;                        // local row 0..63
            const float4 x = h4[(size_t)(wg_row0 + lr) * (HH / 4) + (k0 >> 2) + quad];
            gs_acc[i] += x.x * wsv.x + x.y * wsv.y + x.z * wsv.z + x.w * wsv.w;
            float4 xt;
            xt.x = x.x * twv.x; xt.y = x.y * twv.y;
            xt.z = x.z * twv.z; xt.w = x.w * twv.w;
            *(float4*)&xs[lr * XS_STRIDE + quad * 4] = xt;     // 16B aligned (68*4=272)
        }
        // --- stage pre-swizzled W1 chunk: 4096 contiguous floats ---
        #pragma unroll
        for (int i = 0; i < 8; ++i) {
            const int f = tid + i * NTHR;                     // float4 index 0..1023
            *(float4*)&wt[f * 4] = w1s4[(size_t)c * (WT_FLOATS / 4) + f];
        }
        __syncthreads();

        // --- WMMA: 16 K-steps of 4, 4 N-tiles ---
        #pragma unroll
        for (int ks = 0; ks < 16; ++ks) {
            // A fragment (16x4 f32): lane holds row (wbase+lm), K = ks*4 + 2*lh + {0,1}
            const v2f a = *(const v2f*)&xs[(wbase + lm) * XS_STRIDE + ks * 4 + 2 * lh];
            #pragma unroll
            for (int n = 0; n < 4; ++n) {
                // B fragment: one conflict-free ds_load_b64 per lane
                const v2f bf = *(const v2f*)&wt[(ks * 4 + n) * 64 + lane * 2];
                acc[n] = __builtin_amdgcn_wmma_f32_16x16x4_f32(
                    false, a, false, bf, (short)0, acc[n], false, false);
            }
        }
        __syncthreads();
    }

    // --- epilogue: e = relu(C + b1) @ W2 + b2 (half-wave shuffle reduce) ---
    float b1v[4], w2v[4];
    #pragma unroll
    for (int n = 0; n < 4; ++n) {
        const int col = n * 16 + lm;
        b1v[n] = b1[col];
        w2v[n] = W2[col];
    }
    const float b2v = b2[0];
    #pragma unroll
    for (int r = 0; r < 8; ++r) {
        float p = 0.f;
        #pragma unroll
        for (int n = 0; n < 4; ++n) {
            const float cc = acc[n][r] + b1v[n];
            p += (cc > 0.f ? cc : 0.f) * w2v[n];
        }
        p += __shfl_xor(p, 8, 32);
        p += __shfl_xor(p, 4, 32);
        p += __shfl_xor(p, 2, 32);
        p += __shfl_xor(p, 1, 32);
        if (lm == 0) {
            // lane 0 -> rows wbase+0..7 ; lane 16 -> rows wbase+8..15
            e_out[wg_row0 + wbase + 8 * lh + r] = p + b2v;
        }
    }

    // --- finish gs: reduce the 16 k-quad partials per row ---
    #pragma unroll
    for (int i = 0; i < 8; ++i)
        gsp[(rb + 8 * i) * 17 + quad] = gs_acc[i];
    __syncthreads();
    if (tid < RPW) {
        float s = 0.f;
        #pragma unroll
        for (int q = 0; q < 16; ++q) s += gsp[tid * 17 + q];
        gs_out[wg_row0 + tid] = s;
    }
}

// ---------------------------------------------------------------------------
// Kernel 2: masked online-softmax span scoring. One thread per (b,s).
// ---------------------------------------------------------------------------
__global__ __launch_bounds__(256)
void span_score_kernel(const float* __restrict__ e_pos,
                       const float* __restrict__ gs,
                       const int* __restrict__ seq_lengths,
                       const float* __restrict__ bs,
                       float* __restrict__ out) {
    const int t = blockIdx.x * blockDim.x + threadIdx.x;   // 0..16383
    const int b = t >> 9;
    const int s = t & (SS - 1);
    const int seq = seq_lengths[b];
    const float bsv = bs[0];
    const bool span_valid = (s < seq);

    float m = -INFINITY, den = 0.f, num = 0.f;
    float res[LL];
    #pragma unroll
    for (int l = 0; l < LL; ++l) {
        const int p  = s + l;
        const int pc = (p < SS - 1) ? p : (SS - 1);
        if (p < seq) {
            const float e = e_pos[(b << 9) + pc];
            const float g = gs[(b << 9) + pc];
            if (e > m) {
                const float sc = __expf(m - e);
                den *= sc; num *= sc; m = e;
            }
            const float w = __expf(e - m);
            den += w; num += w * g;
        }
        res[l] = span_valid ? (num / den + bsv) : 0.f;
    }
    float* o = out + (size_t)t * LL;
    #pragma unroll
    for (int l = 0; l < LL; ++l) o[l] = res[l];
}

extern "C" void kernel_launch(void* const* d_in, const int* in_sizes, int n_in,
                              void* d_out, int out_size, void* d_ws, size_t ws_size,
                              hipStream_t stream) {
    (void)in_sizes; (void)n_in; (void)out_size; (void)ws_size;
    const float* hidden = (const float*)d_in[0];
    const int*   seqlen = (const int*)d_in[1];
    const float* tw     = (const float*)d_in[2];
    const float* W1     = (const float*)d_in[3];
    const float* b1     = (const float*)d_in[4];
    const float* W2     = (const float*)d_in[5];
    const float* b2     = (const float*)d_in[6];
    const float* Ws     = (const float*)d_in[7];
    const float* bs     = (const float*)d_in[8];

    float* e_pos = (float*)d_ws;          // 16384 f32
    float* gs    = e_pos + ROWS;          // 16384 f32
    float* w1s   = gs + ROWS;             // 65536 f32 (pre-swizzled W1)

    swizzle_w1_kernel<<<(HH * DD / 2) / 256, 256, 0, stream>>>(W1, w1s);
    span_mlp_wmma_kernel<<<ROWS / RPW, NTHR, 0, stream>>>(
        hidden, tw, w1s, b1, W2, b2, Ws, e_pos, gs);
    span_score_kernel<<<ROWS / 256, 256, 0, stream>>>(
        e_pos, gs, seqlen, bs, (float*)d_out);
}